// MOCCT_50989851738863
// MI455X (gfx1250) — compile-verified
//
#include <hip/hip_runtime.h>
#include <hip/hip_bf16.h>

typedef __attribute__((ext_vector_type(2))) float v2f;
typedef __attribute__((ext_vector_type(8))) float v8f;

namespace {

constexpr int BLOCK = 128;   // 4 waves (wave32)
constexpr int WAVES = 4;
constexpr int ROWS_PER_WAVE = 16;                       // batch rows per wave
constexpr int ROWS_PER_BLOCK = WAVES * ROWS_PER_WAVE;   // 64

// ---- LDS layout (float offsets; all even -> 8B aligned for b64 loads) ----
constexpr int OFF_WQ  = 0;                    // 10x10
constexpr int OFF_WK  = OFF_WQ + 100;
constexpr int OFF_WV  = OFF_WK + 100;
constexpr int OFF_GM  = OFF_WV + 100;         // g_mha pad 16
constexpr int OFF_BM  = OFF_GM + 16;
constexpr int OFF_W1  = OFF_BM + 16;          // [32][32] zero-padded
constexpr int OFF_W2  = OFF_W1 + 32 * 32;
constexpr int OFF_B1  = OFF_W2 + 32 * 32;     // pads of 32
constexpr int OFF_B2  = OFF_B1 + 32;
constexpr int OFF_GP  = OFF_B2 + 32;
constexpr int OFF_BP  = OFF_GP + 32;
constexpr int OFF_W1P = OFF_BP + 32;          // [96][96] hole-padded
constexpr int OFF_W2P = OFF_W1P + 96 * 96;
constexpr int OFF_B1P = OFF_W2P + 96 * 96;    // pads of 96
constexpr int OFF_B2P = OFF_B1P + 96;
constexpr int OFF_GP1 = OFF_B2P + 96;
constexpr int OFF_BP1 = OFF_GP1 + 96;
constexpr int OFF_WC1 = OFF_BP1 + 96;         // [48][96]
constexpr int OFF_BC1 = OFF_WC1 + 48 * 96;
constexpr int OFF_WC2 = OFF_BC1 + 48;         // [16][48]
constexpr int OFF_BC2 = OFF_WC2 + 16 * 48;
constexpr int OFF_T0  = OFF_BC2 + 16;         // per-wave X tiles  [48][32] == [16][96]
constexpr int TILE_F  = 48 * 32;              // 1536 floats = 6 KB
constexpr int OFF_T1  = OFF_T0 + WAVES * TILE_F;
constexpr int SMEM_FLOATS = OFF_T1 + WAVES * TILE_F;   // 39052 floats ~ 153 KB

// ============ WMMA GEMM: Y[16 x NT*16] = X[16 x 4*KSTEPS] * W^T ============
// W stored row-major [N][K] (torch Linear [out,in]) with K padding == STRIDE.
// f32 WMMA 16x16x4 fragment layout (ISA 7.12.2):
//   A 16x4 : lane L, reg j -> (m = L%16, k = k0 + 2*(L/16) + j)
//   B 4x16 : lane L, reg j -> (k = k0 + 2*(L/16) + j, n = L%16)  [B mirrors A, N striped on lanes]
//   C 16x16: lane L, reg r -> (m = r + 8*(L/16), n = L%16)
template<int NT, int KSTEPS, int STRIDE>
__device__ __forceinline__ void wave_gemm(const float* X, const float* W, v8f* C) {
  const int lane = threadIdx.x & 31;
  const int l16  = lane & 15;
  const int koff = (lane >> 4) << 1;      // 0 or 2
  for (int ks = 0; ks < KSTEPS; ++ks) {
    const int k0 = (ks << 2) + koff;
    const v2f a = *(const v2f*)(X + l16 * STRIDE + k0);      // ds_load_b64
#pragma unroll
    for (int nt = 0; nt < NT; ++nt) {
      const v2f b = *(const v2f*)(W + (nt * 16 + l16) * STRIDE + k0);
      C[nt] = __builtin_amdgcn_wmma_f32_16x16x4_f32(
          false, a, false, b, (short)0, C[nt], false, false);
    }
  }
}

template<int NT>
__device__ __forceinline__ void cd_zero(v8f* C) {
#pragma unroll
  for (int nt = 0; nt < NT; ++nt)
#pragma unroll
    for (int r = 0; r < 8; ++r) C[nt][r] = 0.0f;
}

template<int NT>
__device__ __forceinline__ void cd_bias(v8f* C, const float* bias) {
  const int l16 = threadIdx.x & 15;
#pragma unroll
  for (int nt = 0; nt < NT; ++nt) {
    const float bb = bias[nt * 16 + l16];
#pragma unroll
    for (int r = 0; r < 8; ++r) C[nt][r] += bb;
  }
}

template<int NT>
__device__ __forceinline__ void cd_bias_relu(v8f* C, const float* bias) {
  const int l16 = threadIdx.x & 15;
#pragma unroll
  for (int nt = 0; nt < NT; ++nt) {
    const float bb = bias[nt * 16 + l16];
#pragma unroll
    for (int r = 0; r < 8; ++r) C[nt][r] = fmaxf(C[nt][r] + bb, 0.0f);
  }
}

template<int NT, int STRIDE>
__device__ __forceinline__ void cd_store(const v8f* C, float* X) {
  const int lane = threadIdx.x & 31;
  const int l16 = lane & 15, h = lane >> 4;
#pragma unroll
  for (int nt = 0; nt < NT; ++nt)
#pragma unroll
    for (int r = 0; r < 8; ++r)
      X[(r + 8 * h) * STRIDE + nt * 16 + l16] = C[nt][r];
}

template<int NT, int STRIDE>
__device__ __forceinline__ void cd_load_add(v8f* C, const float* X) {
  const int lane = threadIdx.x & 31;
  const int l16 = lane & 15, h = lane >> 4;
#pragma unroll
  for (int nt = 0; nt < NT; ++nt)
#pragma unroll
    for (int r = 0; r < 8; ++r)
      C[nt][r] += X[(r + 8 * h) * STRIDE + nt * 16 + l16];
}

// LayerNorm over valid features (holes are exactly 0 and g/b pads are 0, so
// summing all NT*16 columns equals summing the valid ones).
template<int NT>
__device__ __forceinline__ void cd_layernorm(v8f* C, const float* sG, const float* sB,
                                             float invN) {
  const int l16 = threadIdx.x & 15;
  float g[NT], bb[NT];
#pragma unroll
  for (int nt = 0; nt < NT; ++nt) { g[nt] = sG[nt * 16 + l16]; bb[nt] = sB[nt * 16 + l16]; }
#pragma unroll
  for (int r = 0; r < 8; ++r) {
    float s = 0.f, s2 = 0.f;
#pragma unroll
    for (int nt = 0; nt < NT; ++nt) { const float x = C[nt][r]; s += x; s2 = fmaf(x, x, s2); }
#pragma unroll
    for (int m = 1; m < 16; m <<= 1) {   // stays inside 16-lane half (row owner)
      s  += __shfl_xor(s,  m, 32);
      s2 += __shfl_xor(s2, m, 32);
    }
    const float mean = s * invN;
    const float var  = s2 * invN - mean * mean;
    const float rs   = rsqrtf(var + 1e-6f);
#pragma unroll
    for (int nt = 0; nt < NT; ++nt)
      C[nt][r] = (C[nt][r] - mean) * rs * g[nt] + bb[nt];
  }
}

// PFF: X <- LN( relu(X W1^T + b1) W2^T + b2 + X )
template<int NT, int KSTEPS, int STRIDE>
__device__ __forceinline__ void wave_pff(float* X, float* H,
                                         const float* W1s, const float* B1s,
                                         const float* W2s, const float* B2s,
                                         const float* Gs, const float* Bs, float invN) {
  v8f C[NT];
  cd_zero<NT>(C);
  wave_gemm<NT, KSTEPS, STRIDE>(X, W1s, C);
  cd_bias_relu<NT>(C, B1s);
  cd_store<NT, STRIDE>(C, H);
  v8f D[NT];
  cd_zero<NT>(D);
  wave_gemm<NT, KSTEPS, STRIDE>(H, W2s, D);
  cd_bias<NT>(D, B2s);
  cd_load_add<NT, STRIDE>(D, X);     // residual
  cd_layernorm<NT>(D, Gs, Bs, invN);
  cd_store<NT, STRIDE>(D, X);
}

// ============ tiny MHA (one branch-row = [3,10]) in VALU, no softmax ============
__device__ __forceinline__ void mha_row(const float* q, const float* k, const float* v,
                                        const float* sWq, const float* sWk, const float* sWv,
                                        const float* sG, const float* sB, float* out) {
  float kh[3][10], vh[3][10];
#pragma unroll
  for (int s = 0; s < 3; ++s)
#pragma unroll
    for (int d = 0; d < 10; ++d) {
      float ak = 0.f, av = 0.f;
#pragma unroll
      for (int e = 0; e < 10; ++e) {
        ak = fmaf(k[s * 10 + e], sWk[d * 10 + e], ak);
        av = fmaf(v[s * 10 + e], sWv[d * 10 + e], av);
      }
      kh[s][d] = ak; vh[s][d] = av;
    }
  const float invT = 0.316227766016837933f;  // 1/sqrt(10)
#pragma unroll
  for (int s = 0; s < 3; ++s) {
    float qh[10];
#pragma unroll
    for (int d = 0; d < 10; ++d) {
      float a = 0.f;
#pragma unroll
      for (int e = 0; e < 10; ++e) a = fmaf(q[s * 10 + e], sWq[d * 10 + e], a);
      qh[d] = a;
    }
    float at[3];
#pragma unroll
    for (int t = 0; t < 3; ++t) {
      float a = 0.f;
#pragma unroll
      for (int d = 0; d < 10; ++d) a = fmaf(qh[d], kh[t][d], a);
      at[t] = a * invT;
    }
    float o[10];
    float mean = 0.f;
#pragma unroll
    for (int d = 0; d < 10; ++d) {
      float x = fmaf(at[0], vh[0][d], fmaf(at[1], vh[1][d], at[2] * vh[2][d]));
      x += q[s * 10 + d];                 // residual = pre-projection q
      o[d] = x; mean += x;
    }
    mean *= 0.1f;
    float var = 0.f;
#pragma unroll
    for (int d = 0; d < 10; ++d) { const float t = o[d] - mean; var = fmaf(t, t, var); }
    var *= 0.1f;
    const float rs = rsqrtf(var + 1e-6f);
#pragma unroll
    for (int d = 0; d < 10; ++d) out[s * 10 + d] = (o[d] - mean) * rs * sG[d] + sB[d];
  }
}

__global__ __launch_bounds__(BLOCK)
void mocct_fused_kernel(const float* __restrict__ Q, const float* __restrict__ K,
                        const float* __restrict__ V,
                        const float* __restrict__ Wq, const float* __restrict__ Wk,
                        const float* __restrict__ Wv,
                        const float* __restrict__ g_mha, const float* __restrict__ b_mha,
                        const float* __restrict__ W1, const float* __restrict__ b1,
                        const float* __restrict__ W2, const float* __restrict__ b2,
                        const float* __restrict__ g_pff, const float* __restrict__ b_pff,
                        const float* __restrict__ W1p, const float* __restrict__ b1p,
                        const float* __restrict__ W2p, const float* __restrict__ b2p,
                        const float* __restrict__ g_pff1, const float* __restrict__ b_pff1,
                        const float* __restrict__ Wc1, const float* __restrict__ bc1,
                        const float* __restrict__ Wc2, const float* __restrict__ bc2,
                        float* __restrict__ out, int Bn) {
  extern __shared__ float smem[];
  const int tid  = threadIdx.x;
  const int wave = tid >> 5;
  const int lane = tid & 31;

  // ---- prefetch this block's q/k/v while we stage weights (global_prefetch_b8)
  {
    const long base  = (long)blockIdx.x * ROWS_PER_BLOCK * 90;
    const long total = (long)Bn * 90;
    for (int off = tid * 16; off < ROWS_PER_BLOCK * 90; off += BLOCK * 16) {
      if (base + off < total) {
        __builtin_prefetch(Q + base + off);
        __builtin_prefetch(K + base + off);
        __builtin_prefetch(V + base + off);
      }
    }
  }

  // ---- cooperative weight staging into LDS (zero-padded, hole numbering) ----
  for (int i = tid; i < 100; i += BLOCK) {
    smem[OFF_WQ + i] = Wq[i]; smem[OFF_WK + i] = Wk[i]; smem[OFF_WV + i] = Wv[i];
  }
  for (int i = tid; i < 16; i += BLOCK) {
    smem[OFF_GM + i] = (i < 10) ? g_mha[i] : 0.f;
    smem[OFF_BM + i] = (i < 10) ? b_mha[i] : 0.f;
    smem[OFF_BC2 + i] = (i < 3) ? bc2[i] : 0.f;
  }
  for (int i = tid; i < 32 * 32; i += BLOCK) {
    const int n = i >> 5, k = i & 31;
    const bool ok = (n < 30) && (k < 30);
    smem[OFF_W1 + i] = ok ? W1[n * 30 + k] : 0.f;
    smem[OFF_W2 + i] = ok ? W2[n * 30 + k] : 0.f;
  }
  for (int i = tid; i < 32; i += BLOCK) {
    const bool ok = i < 30;
    smem[OFF_B1 + i] = ok ? b1[i] : 0.f;
    smem[OFF_B2 + i] = ok ? b2[i] : 0.f;
    smem[OFF_GP + i] = ok ? g_pff[i] : 0.f;
    smem[OFF_BP + i] = ok ? b_pff[i] : 0.f;
  }
  for (int i = tid; i < 96 * 96; i += BLOCK) {
    const int cn = i / 96, ck = i % 96;
    const int jn = cn & 31, jk = ck & 31;
    const bool ok = (jn < 30) && (jk < 30);
    const int fn = (cn >> 5) * 30 + jn, fk = (ck >> 5) * 30 + jk;
    smem[OFF_W1P + i] = ok ? W1p[fn * 90 + fk] : 0.f;
    smem[OFF_W2P + i] = ok ? W2p[fn * 90 + fk] : 0.f;
  }
  for (int i = tid; i < 96; i += BLOCK) {
    const int j = i & 31; const bool ok = j < 30;
    const int f = (i >> 5) * 30 + j;
    smem[OFF_B1P + i] = ok ? b1p[f] : 0.f;
    smem[OFF_B2P + i] = ok ? b2p[f] : 0.f;
    smem[OFF_GP1 + i] = ok ? g_pff1[f] : 0.f;
    smem[OFF_BP1 + i] = ok ? b_pff1[f] : 0.f;
  }
  for (int i = tid; i < 48 * 96; i += BLOCK) {
    const int n = i / 96, ck = i % 96;
    const int jk = ck & 31;
    const bool ok = (n < 45) && (jk < 30);
    smem[OFF_WC1 + i] = ok ? Wc1[n * 90 + (ck >> 5) * 30 + jk] : 0.f;
  }
  for (int i = tid; i < 48; i += BLOCK) smem[OFF_BC1 + i] = (i < 45) ? bc1[i] : 0.f;
  for (int i = tid; i < 16 * 48; i += BLOCK) {
    const int n = i / 48, k = i % 48;
    smem[OFF_WC2 + i] = ((n < 3) && (k < 45)) ? Wc2[n * 45 + k] : 0.f;
  }
  __syncthreads();

  float* T0w = smem + OFF_T0 + wave * TILE_F;   // [48][32] == [16][96]
  float* T1w = smem + OFF_T1 + wave * TILE_F;
  const int b0 = (blockIdx.x * WAVES + wave) * ROWS_PER_WAVE;

  // ---- stage 1: MHA #1 (VALU, one branch-row per lane) -> T0 ----
  for (int iter = 0; iter < 2; ++iter) {
    const int r = iter * 32 + lane;              // branch-row 0..47
    if (r < 48) {
      const int bl = r / 3, br = r % 3;
      int bg = b0 + bl; if (bg > Bn - 1) bg = Bn - 1;
      const long gofs = (long)bg * 90 + br * 30;
      float qr[30], kr[30], vr[30], xo[30];
#pragma unroll
      for (int j = 0; j < 30; ++j) { qr[j] = Q[gofs + j]; kr[j] = K[gofs + j]; vr[j] = V[gofs + j]; }
      mha_row(qr, kr, vr, smem + OFF_WQ, smem + OFF_WK, smem + OFF_WV,
              smem + OFF_GM, smem + OFF_BM, xo);
      float* dst = T0w + r * 32;
#pragma unroll
      for (int j = 0; j < 30; ++j) dst[j] = xo[j];
      dst[30] = 0.f; dst[31] = 0.f;
    }
  }

  // ---- stage 2: PFF d=30 #1 (WMMA f32, 3 m-tiles of 16 branch-rows) ----
#pragma unroll
  for (int mt = 0; mt < 3; ++mt)
    wave_pff<2, 8, 32>(T0w + mt * 512, T1w + mt * 512,
                       smem + OFF_W1, smem + OFF_B1, smem + OFF_W2, smem + OFF_B2,
                       smem + OFF_GP, smem + OFF_BP, 1.f / 30.f);

  // ---- stage 3: MHA #2 (self-attention on x), in place on T0 ----
  for (int iter = 0; iter < 2; ++iter) {
    const int r = iter * 32 + lane;
    if (r < 48) {
      const float* src = T0w + r * 32;
      float xr[30], xo[30];
#pragma unroll
      for (int j = 0; j < 30; ++j) xr[j] = src[j];
      mha_row(xr, xr, xr, smem + OFF_WQ, smem + OFF_WK, smem + OFF_WV,
              smem + OFF_GM, smem + OFF_BM, xo);
      float* dst = T0w + r * 32;
#pragma unroll
      for (int j = 0; j < 30; ++j) dst[j] = xo[j];
      dst[30] = 0.f; dst[31] = 0.f;
    }
  }

  // ---- stage 4: PFF d=30 #2 ----
#pragma unroll
  for (int mt = 0; mt < 3; ++mt)
    wave_pff<2, 8, 32>(T0w + mt * 512, T1w + mt * 512,
                       smem + OFF_W1, smem + OFF_B1, smem + OFF_W2, smem + OFF_B2,
                       smem + OFF_GP, smem + OFF_BP, 1.f / 30.f);

  // ---- stage 5: PFF d=90 (T0 reinterpreted as [16][96], same memory) ----
  wave_pff<6, 24, 96>(T0w, T1w,
                      smem + OFF_W1P, smem + OFF_B1P, smem + OFF_W2P, smem + OFF_B2P,
                      smem + OFF_GP1, smem + OFF_BP1, 1.f / 90.f);

  // ---- stage 6: classifier hidden = relu(out @ Wc1^T + bc1) -> T1 [16][48] ----
  {
    v8f C1[3];
    cd_zero<3>(C1);
    wave_gemm<3, 24, 96>(T0w, smem + OFF_WC1, C1);
    cd_bias_relu<3>(C1, smem + OFF_BC1);
    cd_store<3, 48>(C1, T1w);
  }

  // ---- stage 7: logits = hidden @ Wc2^T + bc2, softmax over 3 ----
  {
    v8f C2[1];
    cd_zero<1>(C2);
    wave_gemm<1, 12, 48>(T1w, smem + OFF_WC2, C2);
    cd_bias<1>(C2, smem + OFF_BC2);
    cd_store<1, 16>(C2, T0w);                    // logits [16][16]
  }
  if (lane < 16) {
    const int bg = b0 + lane;
    if (bg < Bn) {
      const float l0 = T0w[lane * 16 + 0];
      const float l1 = T0w[lane * 16 + 1];
      const float l2 = T0w[lane * 16 + 2];
      const float m  = fmaxf(l0, fmaxf(l1, l2));
      const float e0 = expf(l0 - m), e1 = expf(l1 - m), e2 = expf(l2 - m);
      const float inv = 1.0f / (e0 + e1 + e2);
      out[(long)bg * 3 + 0] = e0 * inv;
      out[(long)bg * 3 + 1] = e1 * inv;
      out[(long)bg * 3 + 2] = e2 * inv;
    }
  }
}

}  // namespace

extern "C" void kernel_launch(void* const* d_in, const int* in_sizes, int n_in,
                              void* d_out, int out_size, void* d_ws, size_t ws_size,
                              hipStream_t stream) {
  const float* Q      = (const float*)d_in[0];
  const float* K      = (const float*)d_in[1];
  const float* V      = (const float*)d_in[2];
  const float* Wq     = (const float*)d_in[3];
  const float* Wk     = (const float*)d_in[4];
  const float* Wv     = (const float*)d_in[5];
  const float* g_mha  = (const float*)d_in[6];
  const float* b_mha  = (const float*)d_in[7];
  const float* W1     = (const float*)d_in[8];
  const float* b1     = (const float*)d_in[9];
  const float* W2     = (const float*)d_in[10];
  const float* b2     = (const float*)d_in[11];
  const float* g_pff  = (const float*)d_in[12];
  const float* b_pff  = (const float*)d_in[13];
  const float* W1p    = (const float*)d_in[14];
  const float* b1p    = (const float*)d_in[15];
  const float* W2p    = (const float*)d_in[16];
  const float* b2p    = (const float*)d_in[17];
  const float* g_pff1 = (const float*)d_in[18];
  const float* b_pff1 = (const float*)d_in[19];
  const float* Wc1    = (const float*)d_in[20];
  const float* bc1    = (const float*)d_in[21];
  const float* Wc2    = (const float*)d_in[22];
  const float* bc2    = (const float*)d_in[23];

  const int Bn = in_sizes[0] / 90;
  const int grid = (Bn + ROWS_PER_BLOCK - 1) / ROWS_PER_BLOCK;
  const size_t shmem = (size_t)SMEM_FLOATS * sizeof(float);

  // Opt in to >64KB dynamic LDS (no-op where not needed); safe during capture.
  (void)hipFuncSetAttribute((const void*)mocct_fused_kernel,
                            hipFuncAttributeMaxDynamicSharedMemorySize, (int)shmem);

  mocct_fused_kernel<<<grid, BLOCK, shmem, stream>>>(
      Q, K, V, Wq, Wk, Wv, g_mha, b_mha, W1, b1, W2, b2, g_pff, b_pff,
      W1p, b1p, W2p, b2p, g_pff1, b_pff1, Wc1, bc1, Wc2, bc2,
      (float*)d_out, Bn);
}